// TemplateImageDecoder_50543175139493
// MI455X (gfx1250) — compile-verified
//
#include <hip/hip_runtime.h>
#include <cmath>

#define BB 64
#define KK 40
#define TH_ 16
#define TW_ 16
#define HH 96
#define WW 96

typedef float v2f __attribute__((ext_vector_type(2)));
typedef float v8f __attribute__((ext_vector_type(8)));

// ---------------------------------------------------------------------------
// Decoder: one block per (b,k). Template sigmoid -> LDS once; each wave
// computes 16x16 pixel tiles. Sample coordinates via V_WMMA_F32_16X16X4_F32:
//   ix(h,w) = 8*t1*ys(h) + (8*t0*xs(w) + 8*t2 + 7.5)   (rank-2 in [ys,1])
// A[M] = [ys(h), 1, 0, 0]; Bx = [[8*t1]*16 ; [8*t0*xs(w)+8*t2+7.5]]; C = 0.
// D layout: VGPR j holds row M = j + 8*(lane>=16), col N = lane&15.
// Bilinear gather is branch-free: clamped LDS loads always execute; OOB
// zeroing is folded into the weights via v_cndmask (no EXEC branches).
// ---------------------------------------------------------------------------
__global__ __launch_bounds__(128)
void TemplateImageDecoder_decode_kernel(const float* __restrict__ poses,
                                        const float* __restrict__ presences,
                                        const float* __restrict__ templates,
                                        const float* __restrict__ temperature_logit,
                                        float* __restrict__ out)
{
    __shared__ float sT[TH_ * TW_];

    const int bk  = blockIdx.x;          // 0 .. B*K-1
    const int b   = bk / KK;
    const int k   = bk - b * KK;
    const int tid = threadIdx.x;

    // sigmoid(template_k) -> LDS (1 KB)
    for (int i = tid; i < TH_ * TW_; i += 128)
        sT[i] = 1.0f / (1.0f + expf(-templates[k * TH_ * TW_ + i]));
    __syncthreads();

    // per-(b,k) scalars
    const float t0 = poses[bk * 6 + 0], t1 = poses[bk * 6 + 1], t2 = poses[bk * 6 + 2];
    const float t3 = poses[bk * 6 + 3], t4 = poses[bk * 6 + 4], t5 = poses[bk * 6 + 5];
    const float temperature = log1pf(expf(temperature_logit[0] + 0.5f)) + 1e-4f;
    const float invTemp = 1.0f / temperature;
    const float logpres = logf(fmaxf(presences[bk], 1e-16f));

    const int lane = tid & 31;
    const int wave = tid >> 5;
    const bool lo  = lane < 16;          // K=0,1 half (lanes 16-31 carry K=2,3 = zeros)
    const int  ln  = lane & 15;

    const size_t S     = (size_t)BB * (KK + 1) * HH * WW;          // stack stride
    const size_t obase = ((size_t)b * (KK + 1) + k) * (size_t)(HH * WW);

    for (int t = wave; t < 36; t += 4) {                           // 6x6 tiles of 16x16
        const int ty = t / 6, tx = t - ty * 6;

        const float ysv = (2.0f * (float)(ty * 16 + ln) + 1.0f) * (1.0f / HH) - 1.0f;
        const float xsv = (2.0f * (float)(tx * 16 + ln) + 1.0f) * (1.0f / WW) - 1.0f;

        v2f A  = { lo ? ysv  : 0.0f, lo ? 1.0f : 0.0f };
        v2f Bx = { lo ? 8.0f * t1 : 0.0f,
                   lo ? (8.0f * t0 * xsv + 8.0f * t2 + 7.5f) : 0.0f };
        v2f By = { lo ? 8.0f * t4 : 0.0f,
                   lo ? (8.0f * t3 * xsv + 8.0f * t5 + 7.5f) : 0.0f };
        v8f cz = {};

        // D = A x B + 0 : 256 sample coordinates per WMMA
        v8f IX = __builtin_amdgcn_wmma_f32_16x16x4_f32(false, A, false, Bx,
                                                       (short)0, cz, false, false);
        v8f IY = __builtin_amdgcn_wmma_f32_16x16x4_f32(false, A, false, By,
                                                       (short)0, cz, false, false);

        const int w     = tx * 16 + ln;
        const int hbase = ty * 16 + (lane >> 4) * 8;

        #pragma unroll
        for (int j = 0; j < 8; ++j) {
            const float ix = IX[j], iy = IY[j];
            const float fx0 = floorf(ix), fy0 = floorf(iy);
            const int   x0 = (int)fx0,   y0 = (int)fy0;
            const int   x1 = x0 + 1,     y1 = y0 + 1;
            const float wx1 = ix - fx0,  wy1 = iy - fy0;

            // OOB zeroing folded into the weights (v_cndmask, no branches)
            const float mx0 = ((unsigned)x0 < 16u) ? (1.0f - wx1) : 0.0f;
            const float mx1 = ((unsigned)x1 < 16u) ? wx1          : 0.0f;
            const float my0 = ((unsigned)y0 < 16u) ? (1.0f - wy1) : 0.0f;
            const float my1 = ((unsigned)y1 < 16u) ? wy1          : 0.0f;

            // clamped indices are always in-range: unconditional LDS loads
            const int cx0 = min(max(x0, 0), 15), cx1 = min(max(x1, 0), 15);
            const int cy0 = min(max(y0, 0), 15), cy1 = min(max(y1, 0), 15);
            const float v00 = sT[cy0 * 16 + cx0];
            const float v01 = sT[cy0 * 16 + cx1];
            const float v10 = sT[cy1 * 16 + cx0];
            const float v11 = sT[cy1 * 16 + cx1];

            const float val = my0 * (v00 * mx0 + v01 * mx1)
                            + my1 * (v10 * mx0 + v11 * mx1);

            const size_t addr = obase + (size_t)(hbase + j) * WW + (size_t)w;
            // streaming NT stores: output (193 MB) exceeds L2 and is write-once
            __builtin_nontemporal_store(val * invTemp + logpres, out + addr);      // logits
            __builtin_nontemporal_store(val,                     out + S + addr);  // means
        }
    }
}

// ---------------------------------------------------------------------------
// Background component (slot K) for both stacked outputs.
// ---------------------------------------------------------------------------
__global__ __launch_bounds__(256)
void TemplateImageDecoder_bg_kernel(const float* __restrict__ bg_value,
                                    const float* __restrict__ temperature_logit,
                                    float* __restrict__ out)
{
    const size_t N = (size_t)BB * HH * WW;
    size_t idx = (size_t)blockIdx.x * blockDim.x + threadIdx.x;
    if (idx >= N) return;

    const float bgv = 1.0f / (1.0f + expf(-bg_value[0]));
    const float temperature = log1pf(expf(temperature_logit[0] + 0.5f)) + 1e-4f;
    const float lg = bgv / temperature;

    const size_t bi = idx / (size_t)(HH * WW);
    const size_t p  = idx - bi * (size_t)(HH * WW);
    const size_t S  = (size_t)BB * (KK + 1) * HH * WW;
    const size_t addr = (bi * (KK + 1) + KK) * (size_t)(HH * WW) + p;

    __builtin_nontemporal_store(lg,  out + addr);      // bg logits
    __builtin_nontemporal_store(bgv, out + S + addr);  // bg means
}

extern "C" void kernel_launch(void* const* d_in, const int* in_sizes, int n_in,
                              void* d_out, int out_size, void* d_ws, size_t ws_size,
                              hipStream_t stream)
{
    const float* poses             = (const float*)d_in[0];
    const float* presences         = (const float*)d_in[1];
    const float* templates         = (const float*)d_in[2];
    const float* bg_value          = (const float*)d_in[3];
    const float* temperature_logit = (const float*)d_in[4];
    float* out = (float*)d_out;

    TemplateImageDecoder_decode_kernel<<<BB * KK, 128, 0, stream>>>(
        poses, presences, templates, temperature_logit, out);

    const int N = BB * HH * WW;
    TemplateImageDecoder_bg_kernel<<<(N + 255) / 256, 256, 0, stream>>>(
        bg_value, temperature_logit, out);
}